// ClusterGCN_22686017258280
// MI455X (gfx1250) — compile-verified
//
#include <hip/hip_runtime.h>
#include <math.h>

typedef __attribute__((ext_vector_type(2))) float v2f;
typedef __attribute__((ext_vector_type(8))) float v8f;

#define WMMA_F32(a, b, c) \
    __builtin_amdgcn_wmma_f32_16x16x4_f32(false, (a), false, (b), (short)0, (c), false, false)

// Guaranteed hardware float atomic add (GLOBAL_ATOMIC_ADD_F32, no-return form).
// Avoids any possibility of a CAS-loop expansion on the hot scatter path.
__device__ __forceinline__ void atomic_add_f32(float* p, float v) {
    asm volatile("global_atomic_add_f32 %0, %1, off"
                 :
                 : "v"(p), "v"(v)
                 : "memory");
}

// ---------------- utility kernels ----------------

__global__ void zero_f32_kernel(float* __restrict__ p, long long n) {
    long long i4 = (blockIdx.x * 256LL + threadIdx.x) * 4;
    if (i4 + 3 < n) {
        *(float4*)(p + i4) = make_float4(0.f, 0.f, 0.f, 0.f);
    } else {
        for (long long j = i4; j < n; ++j) p[j] = 0.f;
    }
}

__global__ void count_deg_kernel(const int* __restrict__ dst, float* __restrict__ deg, int E) {
    int e = blockIdx.x * 256 + threadIdx.x;
    if (e < E) atomic_add_f32(&deg[dst[e]], 1.0f);
}

__global__ void finalize_deg_kernel(float* __restrict__ deg, int N) {
    int i = blockIdx.x * 256 + threadIdx.x;
    if (i < N) {
        float d = deg[i];
        deg[i] = (d > 0.f) ? (1.0f / d) : 0.0f;
    }
}

// gather x[src] and scatter-add into agg[dst]; 16 threads per edge, 4 floats each
__global__ void scatter_kernel(const float* __restrict__ x, const int* __restrict__ src,
                               const int* __restrict__ dst, float* __restrict__ agg, int E) {
    long long gid = blockIdx.x * 256LL + threadIdx.x;
    int e = (int)(gid >> 4);
    if (e >= E) return;
    int c = ((int)gid & 15) * 4;
    int s = src[e], d = dst[e];
    float4 v = *(const float4*)(x + (long long)s * 64 + c);
    float* a = agg + (long long)d * 64 + c;
    atomic_add_f32(a + 0, v.x);
    atomic_add_f32(a + 1, v.y);
    atomic_add_f32(a + 2, v.z);
    atomic_add_f32(a + 3, v.w);
}

// ---------------- inProj: y = x[N,128] @ W[64,128]^T + b ----------------
// Writes inp (pre-relu) and x0 = relu(y), both stride 64.
__global__ __launch_bounds__(256) void inproj_kernel(const float* __restrict__ x,
                                                     const float* __restrict__ W,
                                                     const float* __restrict__ b,
                                                     float* __restrict__ inp,
                                                     float* __restrict__ x0, int N) {
    const int K = 128;
    int lane = threadIdx.x & 31;
    int wave = threadIdx.x >> 5;
    int m0 = (blockIdx.x * 8 + wave) * 16;
    if (m0 >= N) return;                 // wave-uniform: EXEC stays all-1 for WMMA
    int lm = lane & 15;
    int kh = (lane >> 4) * 2;            // 0 or 2
    int mrow = m0 + lm;
    int mld = (mrow < N) ? mrow : (N - 1);
    const float* arow = x + (long long)mld * K;

    v8f c0 = {}, c1 = {}, c2 = {}, c3 = {};
    for (int k0 = 0; k0 < K; k0 += 4) {
        int kk = k0 + kh;
        v2f a;
        a.x = arow[kk];
        a.y = arow[kk + 1];
        v2f b0, b1, b2, b3;
        b0.x = W[(0 * 16 + lm) * K + kk]; b0.y = W[(0 * 16 + lm) * K + kk + 1];
        b1.x = W[(1 * 16 + lm) * K + kk]; b1.y = W[(1 * 16 + lm) * K + kk + 1];
        b2.x = W[(2 * 16 + lm) * K + kk]; b2.y = W[(2 * 16 + lm) * K + kk + 1];
        b3.x = W[(3 * 16 + lm) * K + kk]; b3.y = W[(3 * 16 + lm) * K + kk + 1];
        c0 = WMMA_F32(a, b0, c0);
        c1 = WMMA_F32(a, b1, c1);
        c2 = WMMA_F32(a, b2, c2);
        c3 = WMMA_F32(a, b3, c3);
    }

    int rbase = m0 + (lane >> 4) * 8;
#pragma unroll
    for (int v = 0; v < 8; ++v) {
        int r = rbase + v;
        if (r < N) {
            float vals[4] = {c0[v], c1[v], c2[v], c3[v]};
#pragma unroll
            for (int t = 0; t < 4; ++t) {
                int col = t * 16 + lm;
                float y = vals[t] + b[col];
                long long idx = (long long)r * 64 + col;
                inp[idx] = y;
                x0[idx] = fmaxf(y, 0.0f);
            }
        }
    }
}

// ---------------- SAGEConv GEMM ----------------
// out = (agg * deg_inv) @ Wl^T + bl + x @ Wr^T  (K=64), fused epilogue.
// NT = number of 16-wide column tiles (4 for 64 cols, 3 for 47).
// Output written with stride 64 (row-local: safe to alias out == agg).
template <int NT, bool RELU_RES>
__global__ __launch_bounds__(256) void sage_kernel(const float* __restrict__ agg,
                                                   const float* __restrict__ xin,
                                                   const float* __restrict__ dinv,
                                                   const float* __restrict__ Wl,
                                                   const float* __restrict__ Wr,
                                                   const float* __restrict__ bl,
                                                   const float* __restrict__ inp,
                                                   float* __restrict__ out,
                                                   int N, int Ncols) {
    const int K = 64;
    int lane = threadIdx.x & 31;
    int wave = threadIdx.x >> 5;
    int m0 = (blockIdx.x * 8 + wave) * 16;
    if (m0 >= N) return;                 // wave-uniform exit
    int lm = lane & 15;
    int kh = (lane >> 4) * 2;
    int mrow = m0 + lm;
    int mld = (mrow < N) ? mrow : (N - 1);
    float di = dinv[mld];
    const float* arow = agg + (long long)mld * K;
    const float* xrow = xin + (long long)mld * K;

    v8f c[NT] = {};

    // pass 1: mean-aggregated neighbors through Wl
    for (int k0 = 0; k0 < K; k0 += 4) {
        int kk = k0 + kh;
        v2f a;
        a.x = arow[kk] * di;
        a.y = arow[kk + 1] * di;
#pragma unroll
        for (int t = 0; t < NT; ++t) {
            int n = t * 16 + lm;
            if (n >= Ncols) n = Ncols - 1;   // clamp weight row (store is guarded)
            v2f bf;
            bf.x = Wl[n * K + kk];
            bf.y = Wl[n * K + kk + 1];
            c[t] = WMMA_F32(a, bf, c[t]);
        }
    }
    // pass 2: root/self features through Wr, accumulate into same C
    for (int k0 = 0; k0 < K; k0 += 4) {
        int kk = k0 + kh;
        v2f a;
        a.x = xrow[kk];
        a.y = xrow[kk + 1];
#pragma unroll
        for (int t = 0; t < NT; ++t) {
            int n = t * 16 + lm;
            if (n >= Ncols) n = Ncols - 1;
            v2f bf;
            bf.x = Wr[n * K + kk];
            bf.y = Wr[n * K + kk + 1];
            c[t] = WMMA_F32(a, bf, c[t]);
        }
    }

    int rbase = m0 + (lane >> 4) * 8;
#pragma unroll
    for (int v = 0; v < 8; ++v) {
        int r = rbase + v;
        if (r < N) {
#pragma unroll
            for (int t = 0; t < NT; ++t) {
                int col = t * 16 + lm;
                if (col < Ncols) {
                    float y = c[t][v] + bl[col];
                    long long idx = (long long)r * 64 + col;
                    if (RELU_RES) y = fmaxf(y, 0.0f) + 0.2f * inp[idx];
                    out[idx] = y;
                }
            }
        }
    }
}

// ---------------- log_softmax over 47 logits (input stride 64) ----------------
__global__ void logsoftmax_kernel(const float* __restrict__ in, float* __restrict__ out, int N) {
    int n = blockIdx.x * 256 + threadIdx.x;
    if (n >= N) return;
    const float* v = in + (long long)n * 64;
    float vals[47];
    float m = -3.402823466e38f;
#pragma unroll
    for (int j = 0; j < 47; ++j) {
        vals[j] = v[j];
        m = fmaxf(m, vals[j]);
    }
    float s = 0.f;
#pragma unroll
    for (int j = 0; j < 47; ++j) s += expf(vals[j] - m);
    float ls = logf(s);
    float* o = out + (long long)n * 47;
#pragma unroll
    for (int j = 0; j < 47; ++j) o[j] = vals[j] - m - ls;
}

// ---------------- host side ----------------

extern "C" void kernel_launch(void* const* d_in, const int* in_sizes, int n_in,
                              void* d_out, int out_size, void* d_ws, size_t ws_size,
                              hipStream_t stream) {
    const float* x_in = (const float*)d_in[0];
    const int* ei     = (const int*)d_in[1];
    const float* W_in = (const float*)d_in[2];
    const float* b_in = (const float*)d_in[3];
    const float* Wl[4] = {(const float*)d_in[4], (const float*)d_in[7],
                          (const float*)d_in[10], (const float*)d_in[13]};
    const float* Wr[4] = {(const float*)d_in[5], (const float*)d_in[8],
                          (const float*)d_in[11], (const float*)d_in[14]};
    const float* bl[4] = {(const float*)d_in[6], (const float*)d_in[9],
                          (const float*)d_in[12], (const float*)d_in[15]};

    int N = in_sizes[0] / 128;
    int E = in_sizes[1] / 2;
    const int* src = ei;
    const int* dst = ei + E;

    float* ws = (float*)d_ws;
    float* deginv = ws;                          // N
    float* inp    = deginv + N;                  // N*64 (pre-relu inProj, residual)
    float* bufA   = inp + (size_t)N * 64;        // N*64
    float* bufB   = bufA + (size_t)N * 64;       // N*64

    long long nfeat = (long long)N * 64;
    int zb_feat = (int)((nfeat / 4 + 255) / 256) + 1;
    int zb_n    = (N / 4 + 255) / 256 + 1;

    // degrees -> deg_inv
    zero_f32_kernel<<<zb_n, 256, 0, stream>>>(deginv, N);
    count_deg_kernel<<<(E + 255) / 256, 256, 0, stream>>>(dst, deginv, E);
    finalize_deg_kernel<<<(N + 255) / 256, 256, 0, stream>>>(deginv, N);

    // input projection: inp = x@W_in^T + b, bufA = relu(inp)
    int tiles = (N + 15) / 16;
    int gblocks = (tiles + 7) / 8;
    inproj_kernel<<<gblocks, 256, 0, stream>>>(x_in, W_in, b_in, inp, bufA, N);

    long long sthreads = (long long)E * 16;
    int sblocks = (int)((sthreads + 255) / 256);

    float* xcur = bufA;
    float* buf  = bufB;
    for (int i = 0; i < 4; ++i) {
        zero_f32_kernel<<<zb_feat, 256, 0, stream>>>(buf, nfeat);
        scatter_kernel<<<sblocks, 256, 0, stream>>>(xcur, src, dst, buf, E);
        if (i < 3) {
            sage_kernel<4, true><<<gblocks, 256, 0, stream>>>(
                buf, xcur, deginv, Wl[i], Wr[i], bl[i], inp, buf, N, 64);
        } else {
            sage_kernel<3, false><<<gblocks, 256, 0, stream>>>(
                buf, xcur, deginv, Wl[i], Wr[i], bl[i], inp, buf, N, 47);
        }
        float* tmp = xcur; xcur = buf; buf = tmp;
    }

    logsoftmax_kernel<<<(N + 255) / 256, 256, 0, stream>>>(xcur, (float*)d_out, N);
}